// ProductionEMDHybridInSARModel_85779086835979
// MI455X (gfx1250) — compile-verified
//
#include <hip/hip_runtime.h>
#include <hip/hip_bf16.h>

typedef float v2f __attribute__((ext_vector_type(2)));
typedef float v8f __attribute__((ext_vector_type(8)));

#define N_STATIONS 30000
#define N_TIME     512
#define N_HARM     4
#define TWO_PI_F   6.28318530717958647692f

// Wave-level tiling:
//   grid.x = 30000/16 = 1875 station blocks, 128 threads = 4 wave32 per block.
//   Wave w handles 16 stations x 128 times (time quarter w), i.e. 8 WMMA tiles.
// Rank-12 synthesis via 3x V_WMMA_F32_16X16X4_F32:
//   A[s, 0..3]  = amp[s,h]*cos(phi[s,h])      B[0..3, t]  = sin(w_h t)
//   A[s, 4..7]  = amp[s,h]*sin(phi[s,h])      B[4..7, t]  = cos(w_h t)
//   A[s, 8..11] = {offset, trend, 0, 0}       B[8..11, t] = {1, t, 0, 0}
// The memory-bound part (emd harmonic sum + smoothed noise) is fused on the
// C-accumulator layout (lane owns 8 station rows at one t). Streamed-once
// buffers (emd, out) use non-temporal cache hints so the reusable streams
// (base_noise taps, station params) stay resident in L0/L2.
__global__ __launch_bounds__(128)
void insar_wmma_kernel(const float* __restrict__ base_noise,
                       const float* __restrict__ constant_offset,
                       const float* __restrict__ linear_trend,
                       const float* __restrict__ emd,
                       const float* __restrict__ amps,
                       const float* __restrict__ phases,
                       const float* __restrict__ periods,
                       const float* __restrict__ noise_amps,
                       float* __restrict__ out)
{
    const int lane = threadIdx.x & 31;
    const int wid  = threadIdx.x >> 5;      // 0..3 -> time quarter
    const int half = lane >> 4;             // 0/1 -> harmonic pair / C-row half
    const int m    = lane & 15;             // row (A) or column (B) in the 16-tile

    const int sBase = (int)blockIdx.x * 16;

    // ---------- A matrix setup (once per wave) ----------
    const int sA = sBase + m;
    const int h0 = half * 2;                // lanes 0-15: h=0,1 ; lanes 16-31: h=2,3
    float amp0 = amps[sA * N_HARM + h0];
    float amp1 = amps[sA * N_HARM + h0 + 1];
    float ph0  = phases[sA * N_HARM + h0];
    float ph1  = phases[sA * N_HARM + h0 + 1];
    float sp0, cp0, sp1, cp1;
    __sincosf(ph0, &sp0, &cp0);
    __sincosf(ph1, &sp1, &cp1);
    v2f a1 = {amp0 * cp0, amp1 * cp1};      // K = 0..3 : amp*cos(phase)
    v2f a2 = {amp0 * sp0, amp1 * sp1};      // K = 4..7 : amp*sin(phase)
    v2f a3 = {0.f, 0.f};                    // K = 8..11: offset, trend, 0, 0
    if (half == 0) { a3.x = constant_offset[sA]; a3.y = linear_trend[sA]; }

    float p0 = fminf(fmaxf(periods[h0],     20.f), 300.f);
    float p1 = fminf(fmaxf(periods[h0 + 1], 20.f), 300.f);
    const float w0 = TWO_PI_F * __builtin_amdgcn_rcpf(p0);  // v_rcp_f32, no div chain
    const float w1 = TWO_PI_F * __builtin_amdgcn_rcpf(p1);

    // |noise_amp| for the 8 output rows this lane owns in the C/D layout
    float na[8];
    #pragma unroll
    for (int r = 0; r < 8; ++r)
        na[r] = fabsf(noise_amps[sBase + half * 8 + r]);

    const int tChunk = wid * 128;

    for (int i = 0; i < 8; ++i) {
        const int tN = tChunk + i * 16 + m; // this lane's time column

        // ---------- B matrix for this tile ----------
        float s0v, c0v, s1v, c1v;
        __sincosf(w0 * (float)tN, &s0v, &c0v);
        __sincosf(w1 * (float)tN, &s1v, &c1v);
        v2f b1 = {s0v, s1v};
        v2f b2 = {c0v, c1v};
        v2f b3 = {0.f, 0.f};
        if (half == 0) { b3.x = 1.0f; b3.y = (float)tN; }

        // ---------- rank-12 synthesis on the matrix pipe ----------
        v8f c = {};
        c = __builtin_amdgcn_wmma_f32_16x16x4_f32(false, a1, false, b1,
                                                  (short)0, c, false, false);
        c = __builtin_amdgcn_wmma_f32_16x16x4_f32(false, a2, false, b2,
                                                  (short)0, c, false, false);
        c = __builtin_amdgcn_wmma_f32_16x16x4_f32(false, a3, false, b3,
                                                  (short)0, c, false, false);

        // Branch-free boundary handling: clamp the tap addresses (always
        // in-bounds -> unconditional loads, no exec-mask branches), then
        // zero-pad via value selects (v_cndmask).
        const int  offL = (tN > 0)          ? -1 : 0;
        const int  offR = (tN < N_TIME - 1) ?  1 : 0;
        const float wL  = (tN > 0)          ? 1.f : 0.f;
        const float wR  = (tN < N_TIME - 1) ? 1.f : 0.f;

        // ---------- fused memory-bound part on the C layout ----------
        #pragma unroll
        for (int r = 0; r < 8; ++r) {
            const int s = sBase + half * 8 + r;

            // emd: streamed exactly once -> non-temporal loads
            const float* ep = emd + (size_t)s * (N_HARM * N_TIME) + tN;
            float e0 = __builtin_nontemporal_load(ep);
            float e1 = __builtin_nontemporal_load(ep + N_TIME);
            float e2 = __builtin_nontemporal_load(ep + 2 * N_TIME);
            float e3 = __builtin_nontemporal_load(ep + 3 * N_TIME);
            float esum = (e0 + e1) + (e2 + e3);

            // base_noise: neighbor taps have reuse -> regular (RT) loads
            const float* np = base_noise + (size_t)s * N_TIME + tN;
            float nm  = np[offL];
            float n0  = np[0];
            float npp = np[offR];
            float sm  = (nm * wL + n0 + npp * wR) * (1.0f / 3.0f);

            // output: written once -> non-temporal store
            __builtin_nontemporal_store(c[r] + esum + na[r] * sm,
                                        out + (size_t)s * N_TIME + tN);
        }

        if (i < 7) {
            // prefetch next tile of the dominant emd stream (global_prefetch_b8)
            const float* pf =
                emd + (size_t)(sBase + half * 8) * (N_HARM * N_TIME) + tN + 16;
            __builtin_prefetch(pf, 0, 0);
        }
    }
}

extern "C" void kernel_launch(void* const* d_in, const int* in_sizes, int n_in,
                              void* d_out, int out_size, void* d_ws, size_t ws_size,
                              hipStream_t stream) {
    (void)in_sizes; (void)n_in; (void)out_size; (void)d_ws; (void)ws_size;
    // setup_inputs() order:
    // 0 time_vector (unused: t is the index), 1 base_noise, 2 constant_offset,
    // 3 linear_trend, 4 emd_seasonal, 5 residual_amplitudes, 6 residual_phases,
    // 7 residual_periods, 8 noise_amplitudes
    const float* base_noise      = (const float*)d_in[1];
    const float* constant_offset = (const float*)d_in[2];
    const float* linear_trend    = (const float*)d_in[3];
    const float* emd_seasonal    = (const float*)d_in[4];
    const float* res_amps        = (const float*)d_in[5];
    const float* res_phases      = (const float*)d_in[6];
    const float* res_periods     = (const float*)d_in[7];
    const float* noise_amps      = (const float*)d_in[8];
    float* out = (float*)d_out;

    dim3 grid(N_STATIONS / 16);   // 1875 station blocks
    dim3 block(128);              // 4 wave32 per block, one time quarter each
    insar_wmma_kernel<<<grid, block, 0, stream>>>(
        base_noise, constant_offset, linear_trend, emd_seasonal,
        res_amps, res_phases, res_periods, noise_amps, out);
}